// MultiStageFIRFilter_49409303773365
// MI455X (gfx1250) — compile-verified
//
#include <hip/hip_runtime.h>

typedef float v2f __attribute__((ext_vector_type(2)));
typedef float v4f __attribute__((ext_vector_type(4)));
typedef float v8f __attribute__((ext_vector_type(8)));

#define BATCH    8
#define NFRAMES  2000
#define PERIOD   80
#define TAPS     50                 // FILTER_ORDER + 1
#define KCHUNKS  13                 // 52 taps padded, K=4 per wmma
#define TAYLOR   20
#define TSAMP    (NFRAMES * PERIOD) // 160000
#define WINV4    33                 // 132 floats = 33 float4: [t0-52, t0+80)
#define WSTRIDE  132

// One wave32 per (batch, frame): Y(80x2) = Xw(80x52) * C(52x2).
// Window staged in LDS as float4 (zero-padded), 5 interleaved accumulator
// chains of 13 v_wmma_f32_16x16x4_f32. Taps are swapped within each K-pair
// so A-fragments are ascending-address pairs -> ds_load_2addr feeds WMMA
// directly with no register swaps.
__global__ __launch_bounds__(128) void fir_stage(
    const float* __restrict__ xa,   // previous Taylor term, (B, T)
    const float* __restrict__ mc,   // (B, NFRAMES, TAPS)
    float* __restrict__ xout,       // this Taylor term, (B, T)
    float* __restrict__ ybuf,       // running sum, (B, T)
    int initY, float inv_a)
{
    __shared__ float smem[4][WSTRIDE];

    const int lane  = threadIdx.x & 31;
    const int wave  = threadIdx.x >> 5;
    const int frame = blockIdx.x * 4 + wave;        // 0 .. 15999 (exact grid)
    const int b  = frame / NFRAMES;
    const int f  = frame - b * NFRAMES;
    const int fn = (f + 1 < NFRAMES) ? (f + 1) : f; // replicate last frame
    const int n  = lane & 15;                       // B/D column, also A row
    const int h  = lane >> 4;                       // lane half
    const int rowBase = b * TSAMP;
    const int t0f = f * PERIOD;

    // ---- stage 132-float window [t0-52, t0+80) via float4, zero-pad t<0 ----
    const float* xrow = xa + rowBase;
    float* sw = smem[wave];
    for (int i = lane; i < WINV4; i += 32) {        // lane 0 does i=32 too
        int sb  = t0f - 52 + 4 * i;                 // 4-aligned, 16B aligned
        int sbc = (sb < 0) ? 0 : sb;                // clamp keeps alignment
        v4f v = *(const v4f*)(xrow + sbc);
#pragma unroll
        for (int j = 0; j < 4; ++j)
            v[j] = (sb + j < 0) ? 0.0f : v[j];      // branchless zero-pad
        *(v4f*)(sw + 4 * i) = v;
    }

    // ---- B fragments: K-slot 2h <-> tap d0+1, K-slot 2h+1 <-> tap d0 ----
    // Column 0 -> frame f coeffs, column 1 -> frame fn coeffs; tap 0 zeroed.
    v2f bfrag[KCHUNKS];
    {
        const float* cp = mc + (b * NFRAMES + (n ? fn : f)) * TAPS;
#pragma unroll
        for (int k4 = 0; k4 < KCHUNKS; ++k4) {
            int d0 = 4 * k4 + 2 * h;
            float c0 = 0.0f, c1 = 0.0f;
            if (n < 2) {
                if (d0 >= 1 && d0 < TAPS) c0 = cp[d0];   // tap d0 (0 if d0==0)
                if (d0 + 1 < TAPS)        c1 = cp[d0 + 1]; // tap d0+1
            }
            bfrag[k4][0] = c1;      // K-slot 2h   = tap d0+1
            bfrag[k4][1] = c0;      // K-slot 2h+1 = tap d0
        }
    }

    __syncthreads();

    // ---- 5 interleaved accumulator chains over 13 K-chunks ----
    v8f acc[5];
#pragma unroll
    for (int mt = 0; mt < 5; ++mt)
        acc[mt] = (v8f){0.f, 0.f, 0.f, 0.f, 0.f, 0.f, 0.f, 0.f};

    // A[m, slot 2h]   = x[sample ai-1]  (tap d0+1)
    // A[m, slot 2h+1] = x[sample ai]    (tap d0),  ai_lds = 52 + m + 16mt - d0
    const int abase = 52 + n - 2 * h;
#pragma unroll
    for (int k4 = 0; k4 < KCHUNKS; ++k4) {
#pragma unroll
        for (int mt = 0; mt < 5; ++mt) {
            int ai = abase + 16 * mt - 4 * k4;      // in [2, 131]
            v2f af;
            af[0] = sw[ai - 1];                     // ascending pair ->
            af[1] = sw[ai];                         // ds_load_2addr order
            acc[mt] = __builtin_amdgcn_wmma_f32_16x16x4_f32(
                false, af, false, bfrag[k4], (short)0, acc[mt], false, false);
        }
    }

    // ---- blend columns with pre-scaled weights + single shfl_xor ----
    // D layout: vgpr r, lane L -> row m = r + 8*(L>>4), col = L&15.
    // lane col0: z=(1-w)*inv_a*Y_A ; lane col1: z=w*inv_a*Y_B ; xn = z + z^1
#pragma unroll
    for (int mt = 0; mt < 5; ++mt) {
        float o[8];
#pragma unroll
        for (int r = 0; r < 8; ++r) {
            float wm = (float)(16 * mt + r + 8 * h) * (1.0f / PERIOD);
            float wl = ((n == 0) ? (1.0f - wm) : wm) * inv_a;
            float z  = acc[mt][r] * wl;
            o[r] = z + __shfl_xor(z, 1, 32);        // valid in lanes 0/16
        }
        if (n == 0) {   // lanes 0 and 16 each own 8 consecutive samples
            int g = rowBase + t0f + 16 * mt + 8 * h;   // 8-sample aligned
            v4f lo = {o[0], o[1], o[2], o[3]};
            v4f hi = {o[4], o[5], o[6], o[7]};
            *(v4f*)(xout + g)     = lo;
            *(v4f*)(xout + g + 4) = hi;
            const float* ysrc = initY ? (xa + g) : (ybuf + g);
            v4f y0 = *(const v4f*)(ysrc);
            v4f y1 = *(const v4f*)(ysrc + 4);
            *(v4f*)(ybuf + g)     = y0 + lo;
            *(v4f*)(ybuf + g + 4) = y1 + hi;
        }
    }
}

// out[b,t] = y[b,t] * exp( lerp(mc[b,f,0], mc[b,fn,0], (t%80)/80) )
// Vectorized x4: a 4-aligned sample group never crosses a frame (4 | 80).
__global__ __launch_bounds__(256) void scale_out(
    const float* __restrict__ ybuf,
    const float* __restrict__ mc,
    float* __restrict__ out)
{
    int i4 = blockIdx.x * blockDim.x + threadIdx.x;
    if (i4 >= (BATCH * TSAMP) / 4) return;
    int i  = i4 * 4;
    int b  = i / TSAMP;
    int t  = i - b * TSAMP;
    int f  = t / PERIOD;
    int fn = (f + 1 < NFRAMES) ? (f + 1) : f;
    float w0  = (float)(t - f * PERIOD) * (1.0f / PERIOD);
    float c0a = mc[(b * NFRAMES + f)  * TAPS];
    float c0b = mc[(b * NFRAMES + fn) * TAPS];
    v4f y = *(const v4f*)(ybuf + i);
    v4f o;
#pragma unroll
    for (int j = 0; j < 4; ++j) {
        float w = w0 + (float)j * (1.0f / PERIOD);
        o[j] = y[j] * expf((1.0f - w) * c0a + w * c0b);
    }
    *(v4f*)(out + i) = o;
}

extern "C" void kernel_launch(void* const* d_in, const int* in_sizes, int n_in,
                              void* d_out, int out_size, void* d_ws, size_t ws_size,
                              hipStream_t stream)
{
    const float* x  = (const float*)d_in[0];   // (B, T) fp32
    const float* mc = (const float*)d_in[1];   // (B, NFRAMES, TAPS) fp32
    float* out = (float*)d_out;

    const long BT = (long)BATCH * TSAMP;       // 1,280,000
    float* ws  = (float*)d_ws;                 // needs 3*BT*4 = 15.36 MB
    float* xb[2] = { ws, ws + BT };            // ping-pong Taylor terms
    float* yb    = ws + 2 * BT;                // running sum

    dim3 block(128);
    dim3 grid((BATCH * NFRAMES) / 4);          // 4000 blocks, 4 waves each

    for (int s = 1; s <= TAYLOR; ++s) {
        const float* in = (s == 1) ? x : xb[(s - 1) & 1];
        float*       o  = xb[s & 1];
        fir_stage<<<grid, block, 0, stream>>>(in, mc, o, yb,
                                              (s == 1) ? 1 : 0,
                                              1.0f / (float)s);
    }

    int total4 = (BATCH * TSAMP) / 4;          // 320000
    scale_out<<<(total4 + 255) / 256, 256, 0, stream>>>(yb, mc, out);
}